// HeteroGNN_49297634623739
// MI455X (gfx1250) — compile-verified
//
#include <hip/hip_runtime.h>

#define NN 100000
#define NE 1000000
#define H  64
#define LL 8

typedef float v2f __attribute__((ext_vector_type(2)));
typedef float v8f __attribute__((ext_vector_type(8)));

// ---------------------------------------------------------------- utilities
__global__ void zero_f(float* __restrict__ p, long n) {
  long i = (long)blockIdx.x * blockDim.x + threadIdx.x;
  long stride = (long)gridDim.x * blockDim.x;
  for (; i < n; i += stride) p[i] = 0.0f;
}

// ------------------------------------------------- effective weight folding
// A_eff = Wsrc @ Wupd[H:2H]   (agg path)
// D_eff = Wdst @ Wupd[0:H]    (x_dst path)
// b_eff = bsrc@Wupd_bot + bdst@Wupd_top + bupd
__global__ void eff_weights(const float* __restrict__ Wsrc, const float* __restrict__ bsrc,
                            const float* __restrict__ Wdst, const float* __restrict__ bdst,
                            const float* __restrict__ Wupd, const float* __restrict__ bupd,
                            float* __restrict__ Aeff, float* __restrict__ Deff,
                            float* __restrict__ beff) {
  int lm = blockIdx.x;                       // 0..5  (= l*3 + m)
  const float* ws = Wsrc + (size_t)lm * H * H;
  const float* wd = Wdst + (size_t)lm * H * H;
  const float* wu = Wupd + (size_t)lm * 2 * H * H;   // [2H][H] row-major
  for (int idx = threadIdx.x; idx < H * H; idx += blockDim.x) {
    int i = idx >> 6, j = idx & 63;
    float sa = 0.0f, sd = 0.0f;
    for (int k = 0; k < H; ++k) {
      sa += ws[i * H + k] * wu[(H + k) * H + j];
      sd += wd[i * H + k] * wu[k * H + j];
    }
    Aeff[(size_t)lm * H * H + idx] = sa;
    Deff[(size_t)lm * H * H + idx] = sd;
  }
  if (threadIdx.x < H) {
    int j = threadIdx.x;
    float s = bupd[lm * H + j];
    for (int k = 0; k < H; ++k)
      s += bsrc[lm * H + k] * wu[(H + k) * H + j] + bdst[lm * H + k] * wu[k * H + j];
    beff[lm * H + j] = s;
  }
}

// ---------------------------------------------------------- graph degree
__global__ void edge_count(const int* __restrict__ edge, float* __restrict__ cnt, int E) {
  int e = blockIdx.x * blockDim.x + threadIdx.x;
  if (e < E) atomicAdd(&cnt[edge[E + e]], 1.0f);
}

__global__ void invert_cnt(float* __restrict__ c, int n) {
  int i = blockIdx.x * blockDim.x + threadIdx.x;
  if (i < n) c[i] = 1.0f / fmaxf(c[i], 1.0f);
}

// -------------------------------------------------- scatter-add of messages
// 64 lanes per edge (one per feature column) -> coalesced row reads,
// global_atomic_add_f32 into the L2-resident agg buffer.
__global__ void scatter_add(const float* __restrict__ xsrc, const int* __restrict__ edge,
                            float* __restrict__ agg, int E) {
  long tid = (long)blockIdx.x * blockDim.x + threadIdx.x;
  int e = (int)(tid >> 6);
  if (e >= E) return;
  int c = (int)(tid & 63);
  int s = edge[e];
  int d = edge[E + e];
  atomicAdd(&agg[(size_t)d * H + c], xsrc[(size_t)s * H + c]);
}

// --------------------------------------------------------- fused conv GEMM
// y[row, :] = beta*y + alpha*( (agg[row,:]*invc[row]) @ Aeff
//                             + xdst[row,:] @ Deff + beff )
// One wave per 16x64 output strip, V_WMMA_F32_16X16X4_F32, K=64 in steps of 4.
// Weights staged in LDS with K-pair interleaving so each lane's B fragment
// (W[k][n], W[k+1][n]) is one aligned ds_load_b64 -> no repack movs.
__global__ void __launch_bounds__(256)
conv_gemm(const float* __restrict__ agg, const float* __restrict__ invc,
          const float* __restrict__ xdst,
          const float* __restrict__ Aeff, const float* __restrict__ Deff,
          const float* __restrict__ beff,
          float* __restrict__ y, float alpha, int beta, int nrows) {
  __shared__ float sA[H * H];   // interleaved: sA[(kpair*H + n)*2 + (k&1)]
  __shared__ float sD[H * H];
  __shared__ float sB[H];
  for (int i = threadIdx.x; i < H * H; i += blockDim.x) {
    int k = i >> 6, n = i & 63;
    int slot = ((k >> 1) * H + n) * 2 + (k & 1);
    sA[slot] = Aeff[i];
    sD[slot] = Deff[i];
  }
  if (threadIdx.x < H) sB[threadIdx.x] = beff[threadIdx.x];
  __syncthreads();

  int wave = threadIdx.x >> 5;
  int lane = threadIdx.x & 31;
  int strip = blockIdx.x * 8 + wave;           // 16 rows per strip
  int row0 = strip * 16;
  if (row0 >= nrows) return;                    // whole-wave uniform exit

  int half = lane >> 4;                         // 0: K pair {0,1}, 1: {2,3}
  int l16  = lane & 15;
  int r    = row0 + l16;                        // A-fragment row for this lane
  float inv = invc[r];
  const float* aggr  = agg  + (size_t)r * H;
  const float* xdstr = xdst + (size_t)r * H;

  v8f acc0 = {}, acc1 = {}, acc2 = {}, acc3 = {};

  for (int k0 = 0; k0 < H; k0 += 4) {
    int ka = k0 + half * 2;
    v2f aA, aD;
    aA.x = aggr[ka] * inv;  aA.y = aggr[ka + 1] * inv;   // mean fused here
    aD.x = xdstr[ka];       aD.y = xdstr[ka + 1];
    int kbp = (k0 >> 1) + half;                 // interleaved k-pair index
#pragma unroll
    for (int nt = 0; nt < 4; ++nt) {
      int n = nt * 16 + l16;
      v2f bA = *(const v2f*)&sA[(kbp * H + n) * 2];
      v2f bD = *(const v2f*)&sD[(kbp * H + n) * 2];
      v8f* accp = nt == 0 ? &acc0 : nt == 1 ? &acc1 : nt == 2 ? &acc2 : &acc3;
      *accp = __builtin_amdgcn_wmma_f32_16x16x4_f32(false, aA, false, bA,
                                                    (short)0, *accp, false, false);
      *accp = __builtin_amdgcn_wmma_f32_16x16x4_f32(false, aD, false, bD,
                                                    (short)0, *accp, false, false);
    }
  }

  v8f accs[4] = {acc0, acc1, acc2, acc3};
#pragma unroll
  for (int nt = 0; nt < 4; ++nt) {
    int cc = nt * 16 + l16;
    float bb = sB[cc];
#pragma unroll
    for (int j = 0; j < 8; ++j) {              // C/D: VGPR j -> rows j / j+8
      int rr = row0 + half * 8 + j;
      size_t off = (size_t)rr * H + cc;
      float v = alpha * (accs[nt][j] + bb);
      if (beta) v += y[off];
      y[off] = v;
    }
  }
}

// ------------------------------------------------------------- batchnorm
#define BN_BLOCKS 512
__global__ void bn_stats1(const float* __restrict__ y, float* __restrict__ part, int n) {
  __shared__ float s1[256], s2[256];
  int c  = threadIdx.x & 63;
  int rg = threadIdx.x >> 6;                    // 0..3
  float s = 0.0f, q = 0.0f;
  for (int r = blockIdx.x * 4 + rg; r < n; r += gridDim.x * 4) {
    float v = y[(size_t)r * H + c];
    s += v; q += v * v;
  }
  s1[threadIdx.x] = s; s2[threadIdx.x] = q;
  __syncthreads();
  if (rg == 0) {
    s = s1[c] + s1[c + 64] + s1[c + 128] + s1[c + 192];
    q = s2[c] + s2[c + 64] + s2[c + 128] + s2[c + 192];
    part[blockIdx.x * 128 + c]      = s;
    part[blockIdx.x * 128 + 64 + c] = q;
  }
}

__global__ void bn_stats2(const float* __restrict__ part, float* __restrict__ stats,
                          int n, int nb) {
  int c = threadIdx.x;
  if (c < H) {
    float s = 0.0f, q = 0.0f;
    for (int b = 0; b < nb; ++b) { s += part[b * 128 + c]; q += part[b * 128 + 64 + c]; }
    float m = s / (float)n;
    float v = q / (float)n - m * m;             // biased variance
    stats[c]     = m;
    stats[H + c] = rsqrtf(v + 1.0f);            // eps = 1.0
  }
}

__global__ void bn_apply_leaky(const float* __restrict__ y, const float* __restrict__ stats,
                               const float* __restrict__ g, const float* __restrict__ b,
                               float* __restrict__ xout, long total) {
  long i = (long)blockIdx.x * blockDim.x + threadIdx.x;
  long stride = (long)gridDim.x * blockDim.x;
  for (; i < total; i += stride) {
    int c = (int)(i & 63);
    float v = (y[i] - stats[c]) * stats[H + c] * g[c] + b[c];
    xout[i] = v >= 0.0f ? v : 0.01f * v;
  }
}

// ------------------------------------------------------------------ heads
__global__ void head_kernel(const float* __restrict__ x, const float* __restrict__ Wp,
                            const float* __restrict__ bp, float* __restrict__ out, int n) {
  int tid = blockIdx.x * blockDim.x + threadIdx.x;
  int node = tid >> 3;
  if (node >= n) return;
  int l = tid & 7;
  float s = bp[l];
  const float* xr = x + (size_t)node * H;
#pragma unroll 8
  for (int k = 0; k < H; ++k) s += xr[k] * Wp[k * LL + l];
  out[(size_t)node * LL + l] = s;
}

// ------------------------------------------------------------------ driver
extern "C" void kernel_launch(void* const* d_in, const int* in_sizes, int n_in,
                              void* d_out, int out_size, void* d_ws, size_t ws_size,
                              hipStream_t stream) {
  const float* x0   = (const float*)d_in[0];
  const float* x1   = (const float*)d_in[1];
  const int*   e0   = (const int*)d_in[2];
  const int*   e1   = (const int*)d_in[3];
  const int*   e2   = (const int*)d_in[4];
  const float* Wsrc = (const float*)d_in[5];
  const float* bsrc = (const float*)d_in[6];
  const float* Wdst = (const float*)d_in[7];
  const float* bdst = (const float*)d_in[8];
  const float* Wupd = (const float*)d_in[9];
  const float* bupd = (const float*)d_in[10];
  const float* gam  = (const float*)d_in[11];
  const float* bet  = (const float*)d_in[12];
  const float* Wp   = (const float*)d_in[13];
  const float* bp   = (const float*)d_in[14];
  float* out = (float*)d_out;

  const size_t NH = (size_t)NN * H;
  float* W     = (float*)d_ws;
  float* agg   = W;
  float* y0    = W + NH;
  float* y1    = W + 2 * NH;
  float* xc0   = W + 3 * NH;
  float* xc1   = W + 4 * NH;
  float* cnt   = W + 5 * NH;          // cnt0 | cnt1 | cnt2  (3*NN), becomes 1/max(cnt,1)
  float* Aeff  = cnt + 3 * NN;        // 6 * 64*64
  float* Deff  = Aeff + 6 * H * H;
  float* beff  = Deff + 6 * H * H;    // 6 * 64
  float* part  = beff + 6 * H;        // BN_BLOCKS * 128
  float* stats = part + BN_BLOCKS * 128;

  dim3 b256(256);
  int gE      = (NE + 255) / 256;
  int gScat   = (int)(((long)NE * H + 255) / 256);
  int gGemm   = (NN / 16 + 7) / 8;    // 782
  int gElem   = 4096;
  int gHead   = (NN * LL + 255) / 256;

  // effective weights (6 x two 64x64x64 folds)
  eff_weights<<<6, b256, 0, stream>>>(Wsrc, bsrc, Wdst, bdst, Wupd, bupd, Aeff, Deff, beff);

  // degrees (identical for both layers) -> 1/max(cnt,1)
  zero_f<<<512, b256, 0, stream>>>(cnt, 3L * NN);
  edge_count<<<gE, b256, 0, stream>>>(e0, cnt, NE);
  edge_count<<<gE, b256, 0, stream>>>(e1, cnt + NN, NE);
  edge_count<<<gE, b256, 0, stream>>>(e2, cnt + 2 * NN, NE);
  invert_cnt<<<(3 * NN + 255) / 256, b256, 0, stream>>>(cnt, 3 * NN);

  for (int l = 0; l < 2; ++l) {
    const float* xs0 = (l == 0) ? x0 : xc0;
    const float* xs1 = (l == 0) ? x1 : xc1;

    // m0: n0 -> n1, writes y1
    zero_f<<<gElem, b256, 0, stream>>>(agg, (long)NH);
    scatter_add<<<gScat, b256, 0, stream>>>(xs0, e0, agg, NE);
    conv_gemm<<<gGemm, b256, 0, stream>>>(agg, cnt, xs1,
        Aeff + (size_t)(l * 3 + 0) * H * H, Deff + (size_t)(l * 3 + 0) * H * H,
        beff + (l * 3 + 0) * H, y1, 1.0f, 0, NN);

    // m1: n1 -> n0, y0 = 0.5 * conv
    zero_f<<<gElem, b256, 0, stream>>>(agg, (long)NH);
    scatter_add<<<gScat, b256, 0, stream>>>(xs1, e1, agg, NE);
    conv_gemm<<<gGemm, b256, 0, stream>>>(agg, cnt + NN, xs0,
        Aeff + (size_t)(l * 3 + 1) * H * H, Deff + (size_t)(l * 3 + 1) * H * H,
        beff + (l * 3 + 1) * H, y0, 0.5f, 0, NN);

    // m2: n0 -> n0, y0 += 0.5 * conv
    zero_f<<<gElem, b256, 0, stream>>>(agg, (long)NH);
    scatter_add<<<gScat, b256, 0, stream>>>(xs0, e2, agg, NE);
    conv_gemm<<<gGemm, b256, 0, stream>>>(agg, cnt + 2 * NN, xs0,
        Aeff + (size_t)(l * 3 + 2) * H * H, Deff + (size_t)(l * 3 + 2) * H * H,
        beff + (l * 3 + 2) * H, y0, 0.5f, 1, NN);

    // BN + leaky: y0 -> xc0 (type 0), y1 -> xc1 (type 1)
    bn_stats1<<<BN_BLOCKS, b256, 0, stream>>>(y0, part, NN);
    bn_stats2<<<1, 64, 0, stream>>>(part, stats, NN, BN_BLOCKS);
    bn_apply_leaky<<<gElem, b256, 0, stream>>>(y0, stats,
        gam + (l * 2 + 0) * H, bet + (l * 2 + 0) * H, xc0, (long)NH);

    bn_stats1<<<BN_BLOCKS, b256, 0, stream>>>(y1, part, NN);
    bn_stats2<<<1, 64, 0, stream>>>(part, stats, NN, BN_BLOCKS);
    bn_apply_leaky<<<gElem, b256, 0, stream>>>(y1, stats,
        gam + (l * 2 + 1) * H, bet + (l * 2 + 1) * H, xc1, (long)NH);
  }

  // heads
  head_kernel<<<gHead, b256, 0, stream>>>(xc0, Wp,          bp,      out,               NN);
  head_kernel<<<gHead, b256, 0, stream>>>(xc1, Wp + H * LL, bp + LL, out + (size_t)NN * LL, NN);
}